// NMT_90632399880196
// MI455X (gfx1250) — compile-verified
//
#include <hip/hip_runtime.h>
#include <hip/hip_bf16.h>

// ---------------------------------------------------------------------------
// NMT forward (encoder BiLSTM + attention decoder + vocab LSE) for gfx1250.
// All matmuls use V_WMMA_F32_16X16X4_F32 (full fp32 precision on matrix pipe).
// Wave tile: 16x64 (one A fragment feeds 4 WMMAs). Operands are explicitly
// staged into distinct temps so the scheduler can clause the loads and use
// partial s_wait_loadcnt instead of serialized load->wait0->wmma chains.
// ---------------------------------------------------------------------------

#define En 256
#define Hn 512
#define H4 2048
#define Bn 64
#define Sn 40
#define Tn 40
#define VTn 32000

typedef __attribute__((ext_vector_type(2))) float v2f;
typedef __attribute__((ext_vector_type(8))) float v8f;

__device__ __forceinline__ v8f wmma4(v2f a, v2f b, v8f c) {
  // D = A(16x4) * B(4x16) + C, fp32. 8 args: neg_a, A, neg_b, B, c_mod, C, reuse_a, reuse_b
  return __builtin_amdgcn_wmma_f32_16x16x4_f32(false, a, false, b, (short)0, c, false, false);
}

__device__ __forceinline__ float sigm(float x) { return 1.0f / (1.0f + expf(-x)); }

// ------------------------- generic WMMA GEMM: C = X * W^T (+Cin|bias, act) --
// X: [M,K] ldx ; W: [N,K] ldw ; Cin optional init (ldci) ; bias optional [N]
// Each wave: 16x64 strip (4 accumulators, A fragment reused 4x).
// grid = (N/256, M/16), block = 128 (4 waves -> 16 x 256 per block)
__global__ void gemm_xwT(const float* __restrict__ X, int ldx,
                         const float* __restrict__ W, int ldw,
                         const float* __restrict__ Cin, int ldci,
                         const float* __restrict__ bias,
                         float* __restrict__ C, int ldc,
                         int K, int act)
{
  const int lane = threadIdx.x & 31;
  const int wave = threadIdx.x >> 5;
  const int half = lane >> 4;
  const int l16  = lane & 15;
  const int m0 = blockIdx.y * 16;
  const int n0 = blockIdx.x * 256 + wave * 64;

  v8f c0 = {}, c1 = {}, c2 = {}, c3 = {};
  if (Cin) {
#pragma unroll
    for (int v = 0; v < 8; ++v) {
      const float* cr = Cin + (size_t)(m0 + v + 8 * half) * ldci + n0 + l16;
      c0[v] = cr[0];
      c1[v] = cr[16];
      c2[v] = cr[32];
      c3[v] = cr[48];
    }
  }

  const float* Xp  = X + (size_t)(m0 + l16) * ldx + 2 * half;       // A row = m0+l16
  const float* Wp0 = W + (size_t)(n0 + l16) * ldw + 2 * half;       // B col tiles
  const float* Wp1 = Wp0 + (size_t)16 * ldw;
  const float* Wp2 = Wp0 + (size_t)32 * ldw;
  const float* Wp3 = Wp0 + (size_t)48 * ldw;

#pragma unroll 4
  for (int k = 0; k < K; k += 4) {
    v2f a  = *(const v2f*)(Xp  + k);
    v2f b0 = *(const v2f*)(Wp0 + k);
    v2f b1 = *(const v2f*)(Wp1 + k);
    v2f b2 = *(const v2f*)(Wp2 + k);
    v2f b3 = *(const v2f*)(Wp3 + k);
    c0 = wmma4(a, b0, c0);
    c1 = wmma4(a, b1, c1);
    c2 = wmma4(a, b2, c2);
    c3 = wmma4(a, b3, c3);
  }

#pragma unroll
  for (int v = 0; v < 8; ++v) {
    float* cw = C + (size_t)(m0 + v + 8 * half) * ldc + n0 + l16;
    float bb0 = bias ? bias[n0 + l16]      : 0.0f;
    float bb1 = bias ? bias[n0 + 16 + l16] : 0.0f;
    float bb2 = bias ? bias[n0 + 32 + l16] : 0.0f;
    float bb3 = bias ? bias[n0 + 48 + l16] : 0.0f;
    float x0 = c0[v] + bb0, x1 = c1[v] + bb1, x2 = c2[v] + bb2, x3 = c3[v] + bb3;
    if (act == 1) { x0 = tanhf(x0); x1 = tanhf(x1); x2 = tanhf(x2); x3 = tanhf(x3); }
    cw[0]  = x0;
    cw[16] = x1;
    cw[32] = x2;
    cw[48] = x3;
  }
}

// ------------------------------------------------------------- zero scratch
__global__ void zero_kernel(float* p, int n) {
  int i = blockIdx.x * blockDim.x + threadIdx.x;
  if (i < n) p[i] = 0.0f;
}

// -------------------------------------------------- source embedding (+rev)
__global__ void embed_src(const int* __restrict__ src, const int* __restrict__ lens,
                          const float* __restrict__ emb,
                          float* __restrict__ X, float* __restrict__ Xr)
{
  int idx = blockIdx.x * blockDim.x + threadIdx.x;           // B*S*E
  int e = idx & (En - 1); int bs = idx >> 8;
  int s = bs % Sn, b = bs / Sn;
  X[idx] = emb[(size_t)src[b * Sn + s] * En + e];
  int ridx = lens[b] - 1 - s;
  Xr[idx] = (ridx >= 0) ? emb[(size_t)src[b * Sn + ridx] * En + e] : 0.0f;
}

__global__ void embed_tgt(const int* __restrict__ tgt, const float* __restrict__ emb,
                          float* __restrict__ Y)
{
  int idx = blockIdx.x * blockDim.x + threadIdx.x;           // B*T*E  rows b*T+t
  int e = idx & (En - 1); int bt = idx >> 8;
  int t = bt % Tn, b = bt / Tn;
  Y[idx] = emb[(size_t)tgt[b * Tn + t] * En + e];
}

// --------------------------------------------------- encoder LSTM gate step
__global__ void enc_lstm_step(const float* __restrict__ G, float* __restrict__ h,
                              float* __restrict__ c, const int* __restrict__ lens,
                              int t, float* __restrict__ out, int ostride)
{
  int idx = blockIdx.x * blockDim.x + threadIdx.x;           // B*H
  int j = idx & (Hn - 1), b = idx >> 9;
  const float* g = G + (size_t)b * H4;
  float cn = sigm(g[Hn + j]) * c[idx] + sigm(g[j]) * tanhf(g[2 * Hn + j]);
  float hn = sigm(g[3 * Hn + j]) * tanhf(cn);
  bool valid = t < lens[b];
  float hv = valid ? hn : h[idx];
  float cv = valid ? cn : c[idx];
  h[idx] = hv; c[idx] = cv;
  out[(size_t)(b * Sn + t) * ostride + j] = valid ? hv : 0.0f;
}

// --------------------------------------------------- decoder LSTM gate step
__global__ void dec_lstm_step(const float* __restrict__ G, float* __restrict__ h,
                              float* __restrict__ c, float* __restrict__ U)
{
  int idx = blockIdx.x * blockDim.x + threadIdx.x;           // B*H
  int j = idx & (Hn - 1), b = idx >> 9;
  const float* g = G + (size_t)b * H4;
  float cn = sigm(g[Hn + j]) * c[idx] + sigm(g[j]) * tanhf(g[2 * Hn + j]);
  float hn = sigm(g[3 * Hn + j]) * tanhf(cn);
  h[idx] = hn; c[idx] = cn;
  U[(size_t)b * (3 * Hn) + j] = hn;                          // U[:, :H] = h
}

// ---------------------- reverse backward outputs into enc_hidden[:, :, H:]
__global__ void reverse_out(const float* __restrict__ outR, const int* __restrict__ lens,
                            float* __restrict__ encH)
{
  int idx = blockIdx.x * blockDim.x + threadIdx.x;           // B*S*H
  int j = idx & (Hn - 1); int bs = idx >> 9;
  int s = bs % Sn, b = bs / Sn;
  int len = lens[b];
  float v = (s < len) ? outR[(size_t)(b * Sn + (len - 1 - s)) * Hn + j] : 0.0f;
  encH[(size_t)(b * Sn + s) * (2 * Hn) + Hn + j] = v;
}

// --------------------------- concat final states [hf,hb] and [cf,cb]
__global__ void concat_states(const float* hf, const float* hb,
                              const float* cf, const float* cb,
                              float* hcat, float* ccat)
{
  int idx = blockIdx.x * blockDim.x + threadIdx.x;           // B*H
  int j = idx & (Hn - 1), b = idx >> 9;
  hcat[(size_t)b * (2 * Hn) + j]      = hf[idx];
  hcat[(size_t)b * (2 * Hn) + Hn + j] = hb[idx];
  ccat[(size_t)b * (2 * Hn) + j]      = cf[idx];
  ccat[(size_t)b * (2 * Hn) + Hn + j] = cb[idx];
}

// -------------------------------------------------------------- attention
// one block per batch element; writes U[:, H:3H] = attention context a
__global__ void attention(const float* __restrict__ h, const float* __restrict__ encP,
                          const float* __restrict__ encH, const int* __restrict__ lens,
                          float* __restrict__ U)
{
  __shared__ float sh_h[Hn];
  __shared__ float sh_e[Sn];
  __shared__ float sh_inv;
  int b = blockIdx.x, tid = threadIdx.x;                     // 256 threads
  sh_h[tid]       = h[(size_t)b * Hn + tid];
  sh_h[256 + tid] = h[(size_t)b * Hn + 256 + tid];
  __syncthreads();

  int wave = tid >> 5, lane = tid & 31;
  for (int s = wave; s < Sn; s += 8) {
    const float* p = encP + (size_t)(b * Sn + s) * Hn;
    float acc = 0.0f;
    for (int j = lane; j < Hn; j += 32) acc += p[j] * sh_h[j];
    for (int off = 16; off > 0; off >>= 1) acc += __shfl_down(acc, off, 32);
    if (lane == 0) {
      bool masked = (s >= 1) && (s < lens[b]);               // faithful to reference
      sh_e[s] = masked ? -3.0e38f : acc;
    }
  }
  __syncthreads();
  if (tid == 0) {
    float M = sh_e[0];
    for (int s = 1; s < Sn; ++s) M = fmaxf(M, sh_e[s]);
    float Ssum = 0.0f;
    for (int s = 0; s < Sn; ++s) { float e = expf(sh_e[s] - M); sh_e[s] = e; Ssum += e; }
    sh_inv = 1.0f / Ssum;
  }
  __syncthreads();
  for (int j = tid; j < 2 * Hn; j += 256) {
    float acc = 0.0f;
    for (int s = 0; s < Sn; ++s) acc += sh_e[s] * encH[(size_t)(b * Sn + s) * (2 * Hn) + j];
    U[(size_t)b * (3 * Hn) + Hn + j] = acc * sh_inv;
  }
}

// -------------------- vocab projection + fused online log-sum-exp ----------
// 16 rows per block; 8 waves sweep 32000 cols, each wave doing two 16x16
// tiles per iteration (A fragment reused: 3 loads / 2 WMMAs).
__global__ void vocab_lse(const float* __restrict__ Odec,   // [T*B, H] rows t*B+b
                          const float* __restrict__ Wvoc,   // [VT, H]
                          const float* __restrict__ bvoc,
                          const int* __restrict__ target,   // [B, T]
                          float* __restrict__ lse, float* __restrict__ tlog)
{
  __shared__ int   s_tgt[16];
  __shared__ float red_m[8 * 32 * 8];
  __shared__ float red_s[8 * 32 * 8];
  const int lane = threadIdx.x & 31;
  const int wave = threadIdx.x >> 5;
  const int half = lane >> 4, l16 = lane & 15;
  const int r0 = blockIdx.x * 16;

  if (threadIdx.x < 16) {
    int r = r0 + threadIdx.x;
    int t = r >> 6, b = r & (Bn - 1);
    s_tgt[threadIdx.x] = target[b * Tn + t];
  }
  __syncthreads();

  float mrun[8], srun[8];
#pragma unroll
  for (int v = 0; v < 8; ++v) { mrun[v] = -3.0e38f; srun[v] = 0.0f; }

  const float* Ap = Odec + (size_t)(r0 + l16) * Hn + 2 * half;
  for (int iter = 0; iter < VTn / 256; ++iter) {             // 125 iterations
    const int n0 = iter * 256 + wave * 32;
    const float* Bp0 = Wvoc + (size_t)(n0 + l16) * Hn + 2 * half;
    const float* Bp1 = Bp0 + (size_t)16 * Hn;
    // prefetch next chunk's Wvoc rows into GL2 (sequential 65MB stream)
    __builtin_prefetch(Bp0 + (size_t)256 * Hn, 0, 1);
    __builtin_prefetch(Bp1 + (size_t)256 * Hn, 0, 1);
    v8f c0 = {}, c1 = {};
#pragma unroll 8
    for (int k = 0; k < Hn; k += 4) {
      v2f a  = *(const v2f*)(Ap + k);
      v2f b0 = *(const v2f*)(Bp0 + k);
      v2f b1 = *(const v2f*)(Bp1 + k);
      c0 = wmma4(a, b0, c0);
      c1 = wmma4(a, b1, c1);
    }
    const int   col0 = n0 + l16,      col1 = n0 + 16 + l16;
    const float bb0  = bvoc[col0],    bb1  = bvoc[col1];
#pragma unroll
    for (int v = 0; v < 8; ++v) {
      int rowl = v + 8 * half;
      float x0 = c0[v] + bb0;
      float x1 = c1[v] + bb1;
      if (col0 == s_tgt[rowl]) tlog[r0 + rowl] = x0;         // exactly one hit per row
      if (col1 == s_tgt[rowl]) tlog[r0 + rowl] = x1;
      if (x0 > mrun[v]) { srun[v] = srun[v] * expf(mrun[v] - x0) + 1.0f; mrun[v] = x0; }
      else              { srun[v] += expf(x0 - mrun[v]); }
      if (x1 > mrun[v]) { srun[v] = srun[v] * expf(mrun[v] - x1) + 1.0f; mrun[v] = x1; }
      else              { srun[v] += expf(x1 - mrun[v]); }
    }
  }

#pragma unroll
  for (int v = 0; v < 8; ++v) {
    int idx = (wave * 32 + lane) * 8 + v;
    red_m[idx] = mrun[v]; red_s[idx] = srun[v];
  }
  __syncthreads();
  if (threadIdx.x < 16) {
    int m = threadIdx.x, hh = m >> 3, v = m & 7;
    float M = -3.0e38f;
    for (int w = 0; w < 8; ++w)
      for (int l = 0; l < 16; ++l)
        M = fmaxf(M, red_m[(w * 32 + hh * 16 + l) * 8 + v]);
    float Ssum = 0.0f;
    for (int w = 0; w < 8; ++w)
      for (int l = 0; l < 16; ++l) {
        int idx = (w * 32 + hh * 16 + l) * 8 + v;
        Ssum += red_s[idx] * expf(red_m[idx] - M);
      }
    lse[r0 + m] = M + logf(Ssum);
  }
}

// ------------------------------------------------------------- mean NLL
__global__ void final_nll(const float* __restrict__ lse, const float* __restrict__ tlog,
                          float* __restrict__ out)
{
  __shared__ float red[256];
  float acc = 0.0f;
  for (int r = threadIdx.x; r < Tn * Bn; r += 256) acc += lse[r] - tlog[r];
  red[threadIdx.x] = acc;
  __syncthreads();
  for (int s = 128; s > 0; s >>= 1) {
    if (threadIdx.x < s) red[threadIdx.x] += red[threadIdx.x + s];
    __syncthreads();
  }
  if (threadIdx.x == 0) out[0] = red[0] / (float)(Tn * Bn);
}

// ===========================================================================
extern "C" void kernel_launch(void* const* d_in, const int* in_sizes, int n_in,
                              void* d_out, int out_size, void* d_ws, size_t ws_size,
                              hipStream_t stream)
{
  const int*   src     = (const int*)d_in[0];
  const int*   tgtp    = (const int*)d_in[1];
  const int*   lens    = (const int*)d_in[2];
  const float* src_emb = (const float*)d_in[3];
  const float* tgt_emb = (const float*)d_in[4];
  const float* Wih_f   = (const float*)d_in[5];
  const float* Whh_f   = (const float*)d_in[6];
  const float* b_f     = (const float*)d_in[7];
  const float* Wih_b   = (const float*)d_in[8];
  const float* Whh_b   = (const float*)d_in[9];
  const float* b_b     = (const float*)d_in[10];
  const float* Wih_d   = (const float*)d_in[11];
  const float* Whh_d   = (const float*)d_in[12];
  const float* b_d     = (const float*)d_in[13];
  const float* Whp     = (const float*)d_in[14];
  const float* bhp     = (const float*)d_in[15];
  const float* Wcp     = (const float*)d_in[16];
  const float* bcp     = (const float*)d_in[17];
  const float* Watt    = (const float*)d_in[18];
  const float* batt    = (const float*)d_in[19];
  const float* Wcomb   = (const float*)d_in[20];
  const float* bcomb   = (const float*)d_in[21];
  const float* Wvoc    = (const float*)d_in[22];
  const float* bvoc    = (const float*)d_in[23];

  float* ws = (float*)d_ws;
  size_t off = 0;
  auto alloc = [&](size_t n) { float* p = ws + off; off += (n + 3) & ~(size_t)3; return p; };

  float* X    = alloc((size_t)Bn * Sn * En);     // [B,S,E]
  float* Xr   = alloc((size_t)Bn * Sn * En);
  float* Y    = alloc((size_t)Bn * Tn * En);     // rows b*T+t
  float* GXf  = alloc((size_t)Bn * Sn * H4);     // precomputed X @ Wih_f^T + b_f
  float* GXb  = alloc((size_t)Bn * Sn * H4);
  float* GY   = alloc((size_t)Bn * Tn * H4);     // Y @ Wih_d[:, :E]^T + b_d
  float* encH = alloc((size_t)Bn * Sn * 2 * Hn); // enc_hidden [B,S,2H]
  float* outR = alloc((size_t)Bn * Sn * Hn);     // backward scan outputs (unreversed)
  float* encP = alloc((size_t)Bn * Sn * Hn);     // enc_proj
  float* Gf   = alloc((size_t)Bn * H4);
  float* Gb   = alloc((size_t)Bn * H4);
  float* Gd   = alloc((size_t)Bn * H4);
  float* zreg = alloc((size_t)5 * Bn * Hn);      // h_f,c_f,h_b,c_b,zeroO — zero-init
  float* h_f = zreg, *c_f = zreg + 32768, *h_b = zreg + 65536, *c_b = zreg + 98304, *zO = zreg + 131072;
  float* h_d  = alloc((size_t)Bn * Hn);
  float* c_d  = alloc((size_t)Bn * Hn);
  float* hcat = alloc((size_t)Bn * 2 * Hn);
  float* ccat = alloc((size_t)Bn * 2 * Hn);
  float* U    = alloc((size_t)Bn * 3 * Hn);
  float* outs = alloc((size_t)Tn * Bn * Hn);     // decoder outputs O, rows t*B+b
  float* lse  = alloc((size_t)Tn * Bn);
  float* tlog = alloc((size_t)Tn * Bn);

  auto gemm = [&](const float* Xp, int ldx, const float* Wp, int ldw,
                  const float* Cin, int ldci, const float* bias,
                  float* C, int ldc, int M, int N, int K, int act) {
    dim3 g(N / 256, M / 16);
    gemm_xwT<<<g, 128, 0, stream>>>(Xp, ldx, Wp, ldw, Cin, ldci, bias, C, ldc, K, act);
  };

  // 0) zero initial states / o0
  zero_kernel<<<(5 * Bn * Hn + 255) / 256, 256, 0, stream>>>(zreg, 5 * Bn * Hn);

  // 1) embeddings
  embed_src<<<(Bn * Sn * En) / 256, 256, 0, stream>>>(src, lens, src_emb, X, Xr);
  embed_tgt<<<(Bn * Tn * En) / 256, 256, 0, stream>>>(tgtp, tgt_emb, Y);

  // 2) hoisted input-side GEMMs (big WMMA GEMMs)
  gemm(X,  En, Wih_f, En, nullptr, 0, b_f, GXf, H4, Bn * Sn, H4, En, 0);
  gemm(Xr, En, Wih_b, En, nullptr, 0, b_b, GXb, H4, Bn * Sn, H4, En, 0);
  gemm(Y,  En, Wih_d, En + Hn, nullptr, 0, b_d, GY, H4, Bn * Tn, H4, En, 0); // first E cols of Wih_d

  // 3) encoder scans (fwd + bwd per step)
  for (int t = 0; t < Sn; ++t) {
    gemm(h_f, Hn, Whh_f, Hn, GXf + (size_t)t * H4, Sn * H4, nullptr, Gf, H4, Bn, H4, Hn, 0);
    enc_lstm_step<<<(Bn * Hn) / 256, 256, 0, stream>>>(Gf, h_f, c_f, lens, t, encH, 2 * Hn);
    gemm(h_b, Hn, Whh_b, Hn, GXb + (size_t)t * H4, Sn * H4, nullptr, Gb, H4, Bn, H4, Hn, 0);
    enc_lstm_step<<<(Bn * Hn) / 256, 256, 0, stream>>>(Gb, h_b, c_b, lens, t, outR, Hn);
  }

  // 4) reverse backward outputs into enc_hidden[:, :, H:]
  reverse_out<<<(Bn * Sn * Hn) / 256, 256, 0, stream>>>(outR, lens, encH);

  // 5) bridge: dec_h, dec_c
  concat_states<<<(Bn * Hn) / 256, 256, 0, stream>>>(h_f, h_b, c_f, c_b, hcat, ccat);
  gemm(hcat, 2 * Hn, Whp, 2 * Hn, nullptr, 0, bhp, h_d, Hn, Bn, Hn, 2 * Hn, 0);
  gemm(ccat, 2 * Hn, Wcp, 2 * Hn, nullptr, 0, bcp, c_d, Hn, Bn, Hn, 2 * Hn, 0);

  // 6) enc_proj = enc_hidden @ Watt^T + batt
  gemm(encH, 2 * Hn, Watt, 2 * Hn, nullptr, 0, batt, encP, Hn, Bn * Sn, Hn, 2 * Hn, 0);

  // 7) decoder loop
  for (int t = 0; t < Tn; ++t) {
    const float* o_prev = (t == 0) ? zO : outs + (size_t)(t - 1) * Bn * Hn;
    // Gd = GY[:,t,:] + o_prev @ Wih_d[:, E:]^T
    gemm(o_prev, Hn, Wih_d + En, En + Hn, GY + (size_t)t * H4, Tn * H4, nullptr, Gd, H4, Bn, H4, Hn, 0);
    // Gd += h_d @ Whh_d^T
    gemm(h_d, Hn, Whh_d, Hn, Gd, H4, nullptr, Gd, H4, Bn, H4, Hn, 0);
    dec_lstm_step<<<(Bn * Hn) / 256, 256, 0, stream>>>(Gd, h_d, c_d, U);
    attention<<<Bn, 256, 0, stream>>>(h_d, encP, encH, lens, U);
    // O = tanh(U @ Wcomb^T + bcomb) -> outs[t]
    gemm(U, 3 * Hn, Wcomb, 3 * Hn, nullptr, 0, bcomb, outs + (size_t)t * Bn * Hn, Hn,
         Bn, Hn, 3 * Hn, 1);
  }

  // 8) vocab projection with fused online log-sum-exp + target gather
  vocab_lse<<<(Tn * Bn) / 16, 256, 0, stream>>>(outs, Wvoc, bvoc, tgtp, lse, tlog);

  // 9) mean NLL
  final_nll<<<1, 256, 0, stream>>>(lse, tlog, (float*)d_out);
}